// CPCModule_52458730553805
// MI455X (gfx1250) — compile-verified
//
#include <hip/hip_runtime.h>
#include <math.h>

// Problem constants (match reference)
#define B_   64
#define T_   128
#define D_   512
#define H_   30
#define TM_  (T_ - H_)                 // 98
#define NTK_ (TM_ * H_)                // 2940 (t,k) pairs
#define INV_TEMP 14.2857142857142857f  // 1 / 0.07

// LDS tiling: K staged in 64-wide chunks, pitch padded to 68 floats so that
// dword bank = (4*row + col) mod 64 -> each wave's b64 fragment read
// touches 64 distinct banks (conflict-free).
#define KCHUNK 64
#define LDSP   (KCHUNK + 4)

typedef __attribute__((ext_vector_type(2))) float v2f;
typedef __attribute__((ext_vector_type(8))) float v8f;

union V8 { v8f v; float f[8]; };

// ---------------------------------------------------------------------------
// Kernel 1: c_proj[r, n] = sum_k c_seq[r, k] * W[k, n] + b[n]
//   rows r = flattened (B,T) = 8192, cols n = 512, K = 512.
//   Grid (128, 8); each block computes a 64x64 tile with 8 wave32s,
//   each wave owns two 16x16 WMMA accumulators.
// ---------------------------------------------------------------------------
__global__ __launch_bounds__(256)
void cpc_proj_kernel(const float* __restrict__ cseq, const float* __restrict__ W,
                     const float* __restrict__ bias, float* __restrict__ cproj)
{
    __shared__ float sA[64 * LDSP];   // c_seq rows  [64 rows][K chunk]
    __shared__ float sW[64 * LDSP];   // W chunk     [K chunk rows][64 n cols]

    const int rowbase = blockIdx.x * 64;   // over B*T
    const int nbase   = blockIdx.y * 64;   // over D

    const int tid   = threadIdx.x;
    const int wave  = tid >> 5;            // wave32
    const int lane  = tid & 31;
    const int lhalf = lane >> 4;           // 0 or 1
    const int l15   = lane & 15;

    const int mt  = wave & 3;              // m-tile 0..3
    const int nt0 = (wave >> 2) * 2;       // first of two n-tiles

    V8 acc[2] = {};                        // two 16x16 f32 accumulators

    const int lr = tid >> 4;               // 0..15 (loader row)
    const int lc = (tid & 15) * 4;         // 0..60 (loader col, float4)

    for (int kc = 0; kc < D_; kc += KCHUNK) {
        #pragma unroll
        for (int rr = 0; rr < 4; ++rr) {
            const int row = lr + rr * 16;  // 0..63
            const float4 a4 = *(const float4*)(cseq + (size_t)(rowbase + row) * D_ + kc + lc);
            *(float4*)&sA[row * LDSP + lc] = a4;
            const float4 w4 = *(const float4*)(W + (size_t)(kc + row) * D_ + nbase + lc);
            *(float4*)&sW[row * LDSP + lc] = w4;
        }
        __syncthreads();

        #pragma unroll
        for (int step = 0; step < KCHUNK / 4; ++step) {
            const int col = step * 4 + 2 * lhalf;       // k offset for this lane-half
            // A frag (16x4 f32): lane holds A[m=l15, col], A[m=l15, col+1]
            const v2f a = *(const v2f*)&sA[(mt * 16 + l15) * LDSP + col];
            #pragma unroll
            for (int tt = 0; tt < 2; ++tt) {
                const int nt = nt0 + tt;
                // B frag (4x16 f32): lane holds W[col, n=l15], W[col+1, n]
                v2f b;
                b.x = sW[col * LDSP + nt * 16 + l15];
                b.y = sW[(col + 1) * LDSP + nt * 16 + l15];
                acc[tt].v = __builtin_amdgcn_wmma_f32_16x16x4_f32(
                    false, a, false, b, (short)0, acc[tt].v, false, false);
            }
        }
        __syncthreads();
    }

    // Epilogue: bias add + store. C/D layout: m = v + 8*lhalf, n = l15.
    #pragma unroll
    for (int tt = 0; tt < 2; ++tt) {
        const int nt  = nt0 + tt;
        const int col = nbase + nt * 16 + l15;
        const float bn = bias[col];
        #pragma unroll
        for (int v = 0; v < 8; ++v) {
            const int m   = v + 8 * lhalf;
            const int row = rowbase + mt * 16 + m;
            cproj[(size_t)row * D_ + col] = acc[tt].f[v] + bn;
        }
    }
}

// ---------------------------------------------------------------------------
// Kernel 2: one block per (t,k). S[i,j] = <c_proj[i,t,:], z[j,t+k+1,:]>/TEMP
// (64x64x512 GEMM), then per-row logsumexp minus diagonal, block-reduced to
// partials[blockIdx.x]. No atomics -> deterministic.
// ---------------------------------------------------------------------------
__global__ __launch_bounds__(256)
void cpc_scores_kernel(const float* __restrict__ z, const float* __restrict__ cproj,
                       float* __restrict__ partials)
{
    __shared__ float sC[64 * LDSP];   // c_proj rows at time t   (A)
    __shared__ float sZ[64 * LDSP];   // z rows at time u (B^T source), reused for reduce

    const int t = blockIdx.x / H_;
    const int k = blockIdx.x % H_;
    const int u = t + k + 1;

    const int tid   = threadIdx.x;
    const int wave  = tid >> 5;
    const int lane  = tid & 31;
    const int lhalf = lane >> 4;
    const int l15   = lane & 15;

    const int mt  = wave & 3;
    const int nt0 = (wave >> 2) * 2;

    V8 acc[2] = {};

    const int lr = tid >> 4;
    const int lc = (tid & 15) * 4;

    for (int kc = 0; kc < D_; kc += KCHUNK) {
        #pragma unroll
        for (int rr = 0; rr < 4; ++rr) {
            const int row = lr + rr * 16;  // batch index
            const float4 c4 = *(const float4*)(cproj + ((size_t)row * T_ + t) * D_ + kc + lc);
            *(float4*)&sC[row * LDSP + lc] = c4;
            const float4 z4 = *(const float4*)(z + ((size_t)row * T_ + u) * D_ + kc + lc);
            *(float4*)&sZ[row * LDSP + lc] = z4;
        }
        __syncthreads();

        #pragma unroll
        for (int step = 0; step < KCHUNK / 4; ++step) {
            const int col = step * 4 + 2 * lhalf;
            const v2f a = *(const v2f*)&sC[(mt * 16 + l15) * LDSP + col];
            #pragma unroll
            for (int tt = 0; tt < 2; ++tt) {
                const int nt = nt0 + tt;
                // B[k,j] = Z[j,k] -> identical gather shape to A, row = n = l15
                const v2f b = *(const v2f*)&sZ[(nt * 16 + l15) * LDSP + col];
                acc[tt].v = __builtin_amdgcn_wmma_f32_16x16x4_f32(
                    false, a, false, b, (short)0, acc[tt].v, false, false);
            }
        }
        __syncthreads();
    }

    // Stash scaled scores into LDS (reuse sC; safe: all reads fenced above).
    #pragma unroll
    for (int tt = 0; tt < 2; ++tt) {
        const int nt = nt0 + tt;
        #pragma unroll
        for (int v = 0; v < 8; ++v) {
            const int m = v + 8 * lhalf;
            sC[(mt * 16 + m) * LDSP + nt * 16 + l15] = acc[tt].f[v] * INV_TEMP;
        }
    }
    __syncthreads();

    // Per-row logsumexp over j minus diagonal; fixed serial order (deterministic).
    float val = 0.0f;
    if (tid < 64) {
        const float* row = &sC[tid * LDSP];
        float mx = row[0];
        for (int j = 1; j < 64; ++j) mx = fmaxf(mx, row[j]);
        float s = 0.0f;
        for (int j = 0; j < 64; ++j) s += __expf(row[j] - mx);
        val = (mx + __logf(s)) - row[tid];
    }
    sZ[tid] = val;
    __syncthreads();
    #pragma unroll
    for (int off = 128; off > 0; off >>= 1) {
        if (tid < off) sZ[tid] += sZ[tid + off];
        __syncthreads();
    }
    if (tid == 0) partials[blockIdx.x] = sZ[0];
}

// ---------------------------------------------------------------------------
// Kernel 3: deterministic final reduction + mean.
// ---------------------------------------------------------------------------
__global__ __launch_bounds__(256)
void cpc_final_reduce(const float* __restrict__ partials, float* __restrict__ out)
{
    __shared__ float red[256];
    float s = 0.0f;
    for (int i = threadIdx.x; i < NTK_; i += 256) s += partials[i];
    red[threadIdx.x] = s;
    __syncthreads();
    #pragma unroll
    for (int off = 128; off > 0; off >>= 1) {
        if (threadIdx.x < off) red[threadIdx.x] += red[threadIdx.x + off];
        __syncthreads();
    }
    if (threadIdx.x == 0) out[0] = red[0] * (1.0f / ((float)NTK_ * (float)B_));
}

// ---------------------------------------------------------------------------
extern "C" void kernel_launch(void* const* d_in, const int* in_sizes, int n_in,
                              void* d_out, int out_size, void* d_ws, size_t ws_size,
                              hipStream_t stream)
{
    (void)in_sizes; (void)n_in; (void)out_size; (void)ws_size;

    const float* z    = (const float*)d_in[0];  // z_seq  (B,T,D)
    const float* cseq = (const float*)d_in[1];  // c_seq  (B,T,D)
    const float* W    = (const float*)d_in[2];  // W_cpc  (D,D)
    const float* bias = (const float*)d_in[3];  // b_cpc  (D,)
    float* out = (float*)d_out;

    float* cproj    = (float*)d_ws;                          // B*T*D floats (16 MB)
    float* partials = cproj + (size_t)B_ * T_ * D_;          // NTK_ floats

    dim3 g1((B_ * T_) / 64, D_ / 64);            // 128 x 8 blocks
    cpc_proj_kernel<<<g1, 256, 0, stream>>>(cseq, W, bias, cproj);

    cpc_scores_kernel<<<NTK_, 256, 0, stream>>>(z, cproj, partials);

    cpc_final_reduce<<<1, 256, 0, stream>>>(partials, out);
}